// HierEncoder_47751446397032
// MI455X (gfx1250) — compile-verified
//
#include <hip/hip_runtime.h>

typedef __attribute__((ext_vector_type(2))) float v2f;
typedef __attribute__((ext_vector_type(8))) float v8f;

#define SRC_DIM 256
#define EMB_DIM 128
#define TGT_DIM 128

// ---------------------------------------------------------------------------
// Kernel 0: zero the scatter accumulators (summed[T*128] ++ cnt[T])
// ---------------------------------------------------------------------------
__global__ void he_zero_kernel(float* __restrict__ p, int n) {
    int i = blockIdx.x * blockDim.x + threadIdx.x;
    if (i < n) p[i] = 0.0f;
}

// ---------------------------------------------------------------------------
// Kernel 0b: inv_norm[i] = 1 / x_norm[i]  (exact IEEE div, hoisted out of the
// GEMM epilogue so the hot kernel only multiplies)
// ---------------------------------------------------------------------------
__global__ void he_rcp_kernel(const float* __restrict__ x_norm,
                              float* __restrict__ inv_norm, int n) {
    int i = blockIdx.x * blockDim.x + threadIdx.x;
    if (i < n) inv_norm[i] = 1.0f / x_norm[i];
}

// ---------------------------------------------------------------------------
// Kernel 1: x[s,:] = (source_feat[s,:] @ embed) * inv_norm[s], s in [0,S)
// Block = 256 threads = 8 waves; block owns a 16-row strip, wave w owns the
// 16x16 tile at columns [w*16, w*16+16). K = SRC_DIM = 256 -> 64 WMMA k-steps.
// f32 WMMA fragment layout (wave32):
//   A (16x4):  lane = {khalf = lane>>4, mrow = lane&15}; VGPR0/1 = K = khalf*2 + {0,1}
//   B (4x16):  lane column = lane&15;                    VGPR0/1 = K = khalf*2 + {0,1}
//   C (16x16): VGPR r -> M = r + 8*khalf, N = lane&15
// ---------------------------------------------------------------------------
__global__ __launch_bounds__(256) void he_gemm1_kernel(
    const float* __restrict__ src_feat,   // [N, 256] (only first S rows used)
    const float* __restrict__ embed,      // [256, 128]
    const float* __restrict__ inv_norm,   // [S]
    float* __restrict__ x,                // [S, 128] out
    int S)
{
    const int lane  = threadIdx.x & 31;
    const int wave  = threadIdx.x >> 5;       // 0..7 -> column tile
    const int m0    = blockIdx.x * 16;
    const int n0    = wave * 16;
    const int mrow  = lane & 15;
    const int khalf = lane >> 4;              // 0 or 1

    const int m = min(m0 + mrow, S - 1);
    const float* __restrict__ arow = src_feat + (size_t)m * SRC_DIM;

    v8f acc = {};
    #pragma unroll 4
    for (int k = 0; k < SRC_DIM; k += 4) {
        const int kk = k + khalf * 2;
        v2f A;
        A.x = arow[kk];
        A.y = arow[kk + 1];
        v2f B;
        B.x = embed[(size_t)kk       * EMB_DIM + n0 + mrow];
        B.y = embed[(size_t)(kk + 1) * EMB_DIM + n0 + mrow];
        acc = __builtin_amdgcn_wmma_f32_16x16x4_f32(
            /*neg_a=*/false, A, /*neg_b=*/false, B,
            /*c_mod=*/(short)0, acc, /*reuse_a=*/false, /*reuse_b=*/false);
    }

    if (m0 + 16 <= S) {
        // full tile: branch-free epilogue, 8 coalesced stores
        #pragma unroll
        for (int r = 0; r < 8; ++r) {
            const int row = m0 + r + 8 * khalf;
            x[(size_t)row * EMB_DIM + n0 + mrow] = acc[r] * inv_norm[row];
        }
    } else {
        #pragma unroll
        for (int r = 0; r < 8; ++r) {
            const int row = m0 + r + 8 * khalf;
            if (row < S)
                x[(size_t)row * EMB_DIM + n0 + mrow] = acc[r] * inv_norm[row];
        }
    }
}

// ---------------------------------------------------------------------------
// Kernel 2: edge scatter. One wave per edge: lane j loads x[src, j*4 .. j*4+3]
// as a float4 (512B contiguous per wave) and atomically accumulates into
// summed[dst]. lane 0 counts the edge. x (51MB) and summed (10MB) are both
// L2-resident (192MB), so the random traffic and the f32 atomics stay on-chip.
// ---------------------------------------------------------------------------
__global__ __launch_bounds__(256) void he_scatter_kernel(
    const float* __restrict__ x,          // [S, 128]
    const int*   __restrict__ src_idx,    // [E]
    const int*   __restrict__ dst_idx,    // [E]
    float* __restrict__ summed,           // [T, 128]
    float* __restrict__ cnt,              // [T]
    int E)
{
    const int gid  = blockIdx.x * blockDim.x + threadIdx.x;
    const int e    = gid >> 5;
    const int lane = gid & 31;
    if (e >= E) return;

    const int s = src_idx[e];
    const int d = dst_idx[e];

    const float4 v = ((const float4*)(x + (size_t)s * EMB_DIM))[lane];
    float* base = summed + (size_t)d * EMB_DIM + lane * 4;
    unsafeAtomicAdd(base + 0, v.x);
    unsafeAtomicAdd(base + 1, v.y);
    unsafeAtomicAdd(base + 2, v.z);
    unsafeAtomicAdd(base + 3, v.w);
    if (lane == 0) unsafeAtomicAdd(cnt + d, 1.0f);
}

// ---------------------------------------------------------------------------
// Kernel 3: fused mean + output GEMM:
//   out = (summed / max(cnt,1)) @ weight        [T,128] x [128,128]
// Same WMMA tiling as gemm1, K = EMB_DIM = 128 -> 32 k-steps. The per-row
// 1/max(cnt,1) scale folds into the A-fragment load (lane's M row is fixed,
// so the division happens exactly once per lane, before the K-loop).
// ---------------------------------------------------------------------------
__global__ __launch_bounds__(256) void he_gemm2_kernel(
    const float* __restrict__ summed,     // [T, 128]
    const float* __restrict__ cnt,        // [T]
    const float* __restrict__ weight,     // [128, 128]
    float* __restrict__ out,              // [T, 128]
    int T)
{
    const int lane  = threadIdx.x & 31;
    const int wave  = threadIdx.x >> 5;
    const int m0    = blockIdx.x * 16;
    const int n0    = wave * 16;
    const int mrow  = lane & 15;
    const int khalf = lane >> 4;

    const int m = min(m0 + mrow, T - 1);
    const float inv = 1.0f / fmaxf(cnt[m], 1.0f);
    const float* __restrict__ arow = summed + (size_t)m * EMB_DIM;

    v8f acc = {};
    #pragma unroll 4
    for (int k = 0; k < EMB_DIM; k += 4) {
        const int kk = k + khalf * 2;
        v2f A;
        A.x = arow[kk]     * inv;
        A.y = arow[kk + 1] * inv;
        v2f B;
        B.x = weight[(size_t)kk       * TGT_DIM + n0 + mrow];
        B.y = weight[(size_t)(kk + 1) * TGT_DIM + n0 + mrow];
        acc = __builtin_amdgcn_wmma_f32_16x16x4_f32(
            false, A, false, B, (short)0, acc, false, false);
    }

    if (m0 + 16 <= T) {
        #pragma unroll
        for (int r = 0; r < 8; ++r) {
            const int row = m0 + r + 8 * khalf;
            out[(size_t)row * TGT_DIM + n0 + mrow] = acc[r];
        }
    } else {
        #pragma unroll
        for (int r = 0; r < 8; ++r) {
            const int row = m0 + r + 8 * khalf;
            if (row < T)
                out[(size_t)row * TGT_DIM + n0 + mrow] = acc[r];
        }
    }
}

// ---------------------------------------------------------------------------
// Host-side launcher (graph-capture safe: kernels only, all on `stream`)
// Inputs (setup_inputs order):
//   0: source_feat [N,256] f32   1: src_idx [E] i32    2: dst_idx [E] i32
//   3: range_list  [T,2]   i32 (unused)                4: x_norm [N] f32
//   5: embed [256,128] f32       6: weight [128,128] f32
// Output: out [T,128] f32
// Workspace: x[S*128] | summed[T*128] | cnt[T] | inv_norm[S]   (~62 MB)
// ---------------------------------------------------------------------------
extern "C" void kernel_launch(void* const* d_in, const int* in_sizes, int n_in,
                              void* d_out, int out_size, void* d_ws, size_t ws_size,
                              hipStream_t stream) {
    const float* source_feat = (const float*)d_in[0];
    const int*   src_idx     = (const int*)d_in[1];
    const int*   dst_idx     = (const int*)d_in[2];
    const float* x_norm      = (const float*)d_in[4];
    const float* embed       = (const float*)d_in[5];
    const float* weight      = (const float*)d_in[6];
    float*       out         = (float*)d_out;

    const int N = in_sizes[0] / SRC_DIM;       // 120000
    const int E = in_sizes[1];                 // 1000000
    const int T = out_size / TGT_DIM;          // 20000
    const int S = N - T;                       // 100000

    float* x        = (float*)d_ws;                  // [S,128]
    float* summed   = x + (size_t)S * EMB_DIM;       // [T,128]
    float* cnt      = summed + (size_t)T * EMB_DIM;  // [T]
    float* inv_norm = cnt + T;                       // [S]

    // 0) zero accumulators (summed ++ cnt are contiguous) + reciprocal norms
    {
        const int n = T * EMB_DIM + T;
        he_zero_kernel<<<(n + 255) / 256, 256, 0, stream>>>(summed, n);
        he_rcp_kernel<<<(S + 255) / 256, 256, 0, stream>>>(x_norm, inv_norm, S);
    }

    // 1) x = (source_feat @ embed) * inv_norm  (only the S source rows)
    he_gemm1_kernel<<<(S + 15) / 16, 256, 0, stream>>>(source_feat, embed, inv_norm, x, S);

    // 2) edge scatter-add (one wave per edge)
    {
        const long long threads = (long long)E * 32;
        const int blocks = (int)((threads + 255) / 256);
        he_scatter_kernel<<<blocks, 256, 0, stream>>>(x, src_idx, dst_idx, summed, cnt, E);
    }

    // 3) out = (summed / max(cnt,1)) @ weight
    he_gemm2_kernel<<<(T + 15) / 16, 256, 0, stream>>>(summed, cnt, weight, out, T);
}